// HeteroLinear_54408645706450
// MI455X (gfx1250) — compile-verified
//
#include <hip/hip_runtime.h>

typedef float v2f __attribute__((ext_vector_type(2)));
typedef float v4f __attribute__((ext_vector_type(4)));
typedef float v8f __attribute__((ext_vector_type(8)));

#define NROWS   65536
#define KDIM    256
#define FOUT    256
#define NTYPES  16
#define MAX_TILES 4112          // N/16 + T
#define LDS_STRIDE 260          // 256 + 4 pad -> conflict-free ds_load_b64

// ---------------- pass 1: zero counters ----------------
__global__ void hl_zero(int* counts) {
    int i = threadIdx.x;
    if (i < NTYPES) counts[i] = 0;
}

// ---------------- pass 2: histogram types ----------------
__global__ void hl_count(const int* __restrict__ tv, int* __restrict__ counts) {
    int n = blockIdx.x * blockDim.x + threadIdx.x;
    if (n < NROWS) atomicAdd(&counts[tv[n] & (NTYPES - 1)], 1);
}

// ---------------- pass 3: scan + tile table (tiny, serial) ----------------
__global__ void hl_scan(const int* __restrict__ counts, int* __restrict__ offsets,
                        int* __restrict__ cursor, int* __restrict__ tile_type,
                        int* __restrict__ tile_start, int* __restrict__ n_tiles) {
    if (threadIdx.x == 0 && blockIdx.x == 0) {
        int off = 0, nt = 0;
        for (int t = 0; t < NTYPES; ++t) {
            offsets[t] = off;
            cursor[t]  = off;
            int c = counts[t];
            for (int s = 0; s < c; s += 16) {
                tile_type[nt]  = t;
                tile_start[nt] = off + s;
                ++nt;
            }
            off += c;
        }
        offsets[NTYPES] = off;
        *n_tiles = nt;
    }
}

// ---------------- pass 4: scatter row indices into buckets ----------------
__global__ void hl_scatter(const int* __restrict__ tv, int* __restrict__ cursor,
                           int* __restrict__ row_idx) {
    int n = blockIdx.x * blockDim.x + threadIdx.x;
    if (n < NROWS) {
        int t = tv[n] & (NTYPES - 1);
        int p = atomicAdd(&cursor[t], 1);
        row_idx[p] = n;
    }
}

// ---------------- pass 5: grouped GEMM via f32 WMMA ----------------
__global__ __launch_bounds__(512) void hl_gemm(
    const float* __restrict__ x, const float* __restrict__ weight,
    const float* __restrict__ bias, const int* __restrict__ row_idx,
    const int* __restrict__ tile_type, const int* __restrict__ tile_start,
    const int* __restrict__ offsets, const int* __restrict__ n_tiles_p,
    float* __restrict__ out) {
    __shared__ float xs[16 * LDS_STRIDE];

    int tile = blockIdx.x;
    if (tile >= *n_tiles_p) return;

    int t     = tile_type[tile];
    int start = tile_start[tile];
    int nrows = offsets[t + 1] - start;
    nrows = nrows < 16 ? nrows : 16;

    int tid  = threadIdx.x;
    int wave = tid >> 5;        // 0..15 : one 16-col output block per wave
    int lane = tid & 31;

    // Gather 16 x-rows into LDS: wave w loads row w, 8 floats/lane.
    {
        int r  = wave;
        int c0 = lane * 8;
        float* dst = xs + r * LDS_STRIDE + c0;
        if (r < nrows) {
            const float* src = x + (size_t)row_idx[start + r] * KDIM + c0;
            *(v4f*)(dst)     = *(const v4f*)(src);
            *(v4f*)(dst + 4) = *(const v4f*)(src + 4);
        } else {
            v4f z = {};
            *(v4f*)(dst)     = z;
            *(v4f*)(dst + 4) = z;
        }
    }
    __syncthreads();

    int half = lane >> 4;       // K-half select (ISA 7.12.2 layout)
    int ln   = lane & 15;       // A: row M ; B/C/D: column N within block
    int ncol = wave * 16 + ln;  // global output column

    const float* wt = weight + (size_t)t * KDIM * FOUT;

    v8f acc = {};
#pragma unroll 8
    for (int kk = 0; kk < KDIM; kk += 4) {
        int ka = kk + 2 * half;
        // A fragment: {x[m=ln][ka], x[m=ln][ka+1]}  (single ds_load_b64)
        v2f a = *(const v2f*)(xs + ln * LDS_STRIDE + ka);
        // B fragment: {W[ka][ncol], W[ka+1][ncol]}
        v2f b;
        b.x = wt[(size_t)ka * FOUT + ncol];
        b.y = wt[(size_t)(ka + 1) * FOUT + ncol];
        acc = __builtin_amdgcn_wmma_f32_16x16x4_f32(
            false, a, false, b, (short)0, acc, false, false);
    }

    float bv = bias[t * FOUT + ncol];
#pragma unroll
    for (int v = 0; v < 8; ++v) {
        int m = v + 8 * half;   // C/D layout: VGPR v <-> row m
        if (m < nrows) {
            int gr = row_idx[start + m];
            out[(size_t)gr * FOUT + ncol] = acc[v] + bv;
        }
    }
}

// ---------------- fallback (ws too small): per-row VALU dot ----------------
__global__ __launch_bounds__(256) void hl_fallback(
    const float* __restrict__ x, const int* __restrict__ tv,
    const float* __restrict__ weight, const float* __restrict__ bias,
    float* __restrict__ out) {
    __shared__ float xrow[KDIM];
    int n = blockIdx.x;
    int t = tv[n] & (NTYPES - 1);
    xrow[threadIdx.x] = x[(size_t)n * KDIM + threadIdx.x];
    __syncthreads();
    const float* wt = weight + (size_t)t * KDIM * FOUT;
    float acc = bias[t * FOUT + threadIdx.x];
    for (int k = 0; k < KDIM; ++k)
        acc = fmaf(xrow[k], wt[(size_t)k * FOUT + threadIdx.x], acc);
    out[(size_t)n * FOUT + threadIdx.x] = acc;
}

extern "C" void kernel_launch(void* const* d_in, const int* in_sizes, int n_in,
                              void* d_out, int out_size, void* d_ws, size_t ws_size,
                              hipStream_t stream) {
    const float* x  = (const float*)d_in[0];
    const int*   tv = (const int*)d_in[1];
    const float* w  = (const float*)d_in[2];
    const float* b  = (const float*)d_in[3];
    float* out = (float*)d_out;

    // workspace layout (bytes)
    const size_t OFF_COUNTS = 0;                       // 16 ints
    const size_t OFF_CURSOR = 64;                      // 16 ints
    const size_t OFF_OFFS   = 128;                     // 17 ints
    const size_t OFF_NT     = 200;                     // 1 int
    const size_t OFF_TTYPE  = 256;                     // MAX_TILES ints
    const size_t OFF_TSTART = OFF_TTYPE + MAX_TILES * 4;
    const size_t OFF_RIDX   = OFF_TSTART + MAX_TILES * 4;
    const size_t NEED       = OFF_RIDX + (size_t)NROWS * 4;

    if (ws_size < NEED) {
        hl_fallback<<<NROWS, 256, 0, stream>>>(x, tv, w, b, out);
        return;
    }

    char* ws = (char*)d_ws;
    int* counts     = (int*)(ws + OFF_COUNTS);
    int* cursor     = (int*)(ws + OFF_CURSOR);
    int* offsets    = (int*)(ws + OFF_OFFS);
    int* n_tiles    = (int*)(ws + OFF_NT);
    int* tile_type  = (int*)(ws + OFF_TTYPE);
    int* tile_start = (int*)(ws + OFF_TSTART);
    int* row_idx    = (int*)(ws + OFF_RIDX);

    hl_zero<<<1, 32, 0, stream>>>(counts);
    hl_count<<<NROWS / 256, 256, 0, stream>>>(tv, counts);
    hl_scan<<<1, 32, 0, stream>>>(counts, offsets, cursor, tile_type, tile_start, n_tiles);
    hl_scatter<<<NROWS / 256, 256, 0, stream>>>(tv, cursor, row_idx);
    hl_gemm<<<MAX_TILES, 512, 0, stream>>>(x, w, b, row_idx, tile_type, tile_start,
                                           offsets, n_tiles, out);
}